// CGM_23862838296583
// MI455X (gfx1250) — compile-verified
//
#include <hip/hip_runtime.h>
#include <hip/hip_bf16.h>

typedef __attribute__((ext_vector_type(2))) float v2f;
typedef __attribute__((ext_vector_type(8))) float v8f;

#define NN 512
#define HH 64
#define NH (NN * HH)
#define TT 6
#define RR 3

// Activation codes for the matmul epilogue
#define ACT_NONE 0
#define ACT_RELU 1
#define ACT_SIG  2
#define ACT_TANH 3

__device__ __forceinline__ float sigf(float x) { return 1.0f / (1.0f + expf(-x)); }

// ---------------------------------------------------------------------------
// C[n, j] = act( sum_k A[n,k] * W[j,k] + bias[j] )   (i.e. A @ W.T + b)
// A: (rows, K) row-major, W: (Jout, K) row-major. One wave per 16x16 tile.
// Uses V_WMMA_F32_16X16X4_F32 (full fp32 precision).
// ---------------------------------------------------------------------------
__global__ void mm_wmma(const float* __restrict__ A, const float* __restrict__ W,
                        const float* __restrict__ bias, float* __restrict__ C,
                        int K, int Jout, int act)
{
    const int m0   = blockIdx.x * 16;
    const int j0   = blockIdx.y * 16;
    const int lane = threadIdx.x;      // 0..31, one full wave32
    const int half = lane >> 4;        // 0 or 1
    const int lr   = lane & 15;

    const float* __restrict__ arow = A + (size_t)(m0 + lr) * K;
    const float* __restrict__ wrow = W + (size_t)(j0 + lr) * K;

    v8f acc = {};
    for (int k0 = 0; k0 < K; k0 += 4) {
        v2f a, b;
        // A 16x4 frag: lane holds row M=lr, ks {0,1}+2*half
        a[0] = arow[k0 + 2 * half + 0];
        a[1] = arow[k0 + 2 * half + 1];
        // B 4x16 frag: lane holds col N=lr (output j), ks {0,1}+2*half
        b[0] = wrow[k0 + 2 * half + 0];
        b[1] = wrow[k0 + 2 * half + 1];
        acc = __builtin_amdgcn_wmma_f32_16x16x4_f32(false, a, false, b,
                                                    (short)0, acc, false, false);
    }

    const float bv = bias ? bias[j0 + lr] : 0.0f;
#pragma unroll
    for (int v = 0; v < 8; ++v) {
        float val = acc[v] + bv;
        if (act == ACT_RELU)      val = fmaxf(val, 0.0f);
        else if (act == ACT_SIG)  val = sigf(val);
        else if (act == ACT_TANH) val = tanhf(val);
        const int m = m0 + v + 8 * half;
        C[(size_t)m * Jout + j0 + lr] = val;
    }
}

// ---------------------------------------------------------------------------
// out[n,o] = tanh( sum_r sum_m adjs[r,n,m] * support[r,m,o] )
// adjs: (R, N, N), support: (R, N, H). One wave per 16x16 output tile.
// ---------------------------------------------------------------------------
__global__ void rgcn_agg_wmma(const float* __restrict__ adjs,
                              const float* __restrict__ support,
                              float* __restrict__ out)
{
    const int m0   = blockIdx.x * 16;  // n
    const int j0   = blockIdx.y * 16;  // o
    const int lane = threadIdx.x;
    const int half = lane >> 4;
    const int lr   = lane & 15;

    v8f acc = {};
    for (int r = 0; r < RR; ++r) {
        const float* __restrict__ Ar = adjs + (size_t)r * NN * NN + (size_t)(m0 + lr) * NN;
        const float* __restrict__ Sr = support + (size_t)r * NH;
        for (int k0 = 0; k0 < NN; k0 += 4) {
            v2f a, b;
            a[0] = Ar[k0 + 2 * half + 0];
            a[1] = Ar[k0 + 2 * half + 1];
            b[0] = Sr[(size_t)(k0 + 2 * half + 0) * HH + j0 + lr];
            b[1] = Sr[(size_t)(k0 + 2 * half + 1) * HH + j0 + lr];
            acc = __builtin_amdgcn_wmma_f32_16x16x4_f32(false, a, false, b,
                                                        (short)0, acc, false, false);
        }
    }
#pragma unroll
    for (int v = 0; v < 8; ++v) {
        const int m = m0 + v + 8 * half;
        out[(size_t)m * HH + j0 + lr] = tanhf(acc[v]);
    }
}

// ---------------------------------------------------------------------------
// Pack helpers: concatenate H-wide feature blocks along the feature dim.
// ---------------------------------------------------------------------------
__global__ void pack5(const float* __restrict__ s0, const float* __restrict__ s1,
                      const float* __restrict__ s2, const float* __restrict__ s3,
                      const float* __restrict__ s4, float* __restrict__ dst, int rows)
{
    const int idx = blockIdx.x * blockDim.x + threadIdx.x;
    if (idx >= rows * 320) return;
    const int n = idx / 320, j = idx % 320;
    const int blk = j >> 6, h = j & 63;
    const float* s = (blk == 0) ? s0 : (blk == 1) ? s1 : (blk == 2) ? s2
                      : (blk == 3) ? s3 : s4;
    dst[idx] = s[n * HH + h];
}

__global__ void pack2(const float* __restrict__ s0, const float* __restrict__ s1,
                      float* __restrict__ dst, int rows)
{
    const int idx = blockIdx.x * blockDim.x + threadIdx.x;
    if (idx >= rows * 128) return;
    const int n = idx / 128, j = idx % 128;
    const float* s = (j < 64) ? s0 : s1;
    dst[idx] = s[n * HH + (j & 63)];
}

__global__ void copyk(const float* __restrict__ src, float* __restrict__ dst, int n)
{
    const int i = blockIdx.x * blockDim.x + threadIdx.x;
    if (i < n) dst[i] = src[i];
}

// ---------------------------------------------------------------------------
// SLSTM pointwise:
// new_c = sig(f)*c + sig(i)*tanh(u) + sig(t)*h_t ; new_h = sig(o)*tanh(new_c)
// gates: (N, 5H) columns [i|f|o|u|t]
// ---------------------------------------------------------------------------
__global__ void slstm_pw(const float* __restrict__ gates, const float* __restrict__ c_in,
                         const float* __restrict__ h_carry,
                         float* __restrict__ h_out, float* __restrict__ c_out)
{
    const int idx = blockIdx.x * blockDim.x + threadIdx.x;
    if (idx >= NH) return;
    const int n = idx >> 6, h = idx & 63;
    const float* g = gates + (size_t)n * 320;
    const float i = sigf(g[h]);
    const float f = sigf(g[64 + h]);
    const float o = sigf(g[128 + h]);
    const float u = tanhf(g[192 + h]);
    const float t = sigf(g[256 + h]);
    const float c = f * c_in[idx] + i * u + t * h_carry[idx];
    c_out[idx] = c;
    h_out[idx] = o * tanhf(c);
}

// GLSTM init pointwise: new_c = sig(f)*cg + sig(i)*tanh(u) ; new_g = o*tanh(new_c)
// (o deliberately NOT sigmoided — faithful to reference)
__global__ void glstm_init_pw(const float* __restrict__ gates, const float* __restrict__ cg_in,
                              float* __restrict__ g_out, float* __restrict__ cg_out)
{
    const int idx = blockIdx.x * blockDim.x + threadIdx.x;
    if (idx >= NH) return;
    const int n = idx >> 6, h = idx & 63;
    const float* g = gates + (size_t)n * 320;
    const float i = sigf(g[h]);
    const float f = sigf(g[64 + h]);
    const float o = g[128 + h];         // raw
    const float u = tanhf(g[192 + h]);
    const float c = f * cg_in[idx] + i * u;
    cg_out[idx] = c;
    g_out[idx]  = o * tanhf(c);
}

// GLSTM pointwise (gates already sigmoided by matmul epilogue):
// new_c = f*cg + weighted ; new_g = o*tanh(new_c)
__global__ void glstm_pw(const float* __restrict__ gates, const float* __restrict__ cg_in,
                         const float* __restrict__ weighted,
                         float* __restrict__ g_out, float* __restrict__ cg_out)
{
    const int idx = blockIdx.x * blockDim.x + threadIdx.x;
    if (idx >= NH) return;
    const int n = idx >> 6, h = idx & 63;
    const float f = gates[(size_t)n * 320 + 64 + h];
    const float o = gates[(size_t)n * 320 + 128 + h];
    const float c = f * cg_in[idx] + weighted[idx];
    cg_out[idx] = c;
    g_out[idx]  = o * tanhf(c);
}

// h_avg[h] = mean_n h[n,h]; bias2[j] = dot(h_avg, U[j,:]) + Ub[j]
__global__ void havg_bias2(const float* __restrict__ h, const float* __restrict__ U,
                           const float* __restrict__ Ub, float* __restrict__ bias2)
{
    __shared__ float havg[HH];
    const int t = threadIdx.x;
    if (t < HH) {
        float s = 0.0f;
        for (int n = 0; n < NN; ++n) s += h[n * HH + t];
        havg[t] = s * (1.0f / (float)NN);
    }
    __syncthreads();
    if (t < 320) {
        const float* urow = U + (size_t)t * HH;
        float s = Ub[t];
        for (int k = 0; k < HH; ++k) s += havg[k] * urow[k];
        bias2[t] = s;
    }
}

// weighted[n,h] = sum_m c[m,h]*exp(sig(a[n,h]+b[m,h])) / sum_m exp(sig(...))
// (identical to softmax over m of sig(a+b), dotted with c; sig in (0,1) so no
//  max-shift needed numerically)
__global__ void fw_kernel(const float* __restrict__ a, const float* __restrict__ b,
                          const float* __restrict__ c, float* __restrict__ weighted)
{
    const int n = blockIdx.x;
    const int h = threadIdx.x;   // 64 threads
    const float an = a[n * HH + h];
    float denom = 0.0f, numer = 0.0f;
    for (int m = 0; m < NN; ++m) {
        const float s = sigf(an + b[m * HH + h]);
        const float e = expf(s);
        denom += e;
        numer += c[m * HH + h] * e;
    }
    weighted[n * HH + h] = numer / denom;
}

// ===========================================================================
// Host orchestration
// ===========================================================================
extern "C" void kernel_launch(void* const* d_in, const int* in_sizes, int n_in,
                              void* d_out, int out_size, void* d_ws, size_t ws_size,
                              hipStream_t stream)
{
    (void)out_size; (void)ws_size;

    // ---- Locate inputs robustly (insertion-order vs sorted-key flattening) ----
    const float *price, *volume, *adjs_p, *adjs_v;
    int par0;
    const bool data_first = (in_sizes[0] == TT * NN * 32);  // price_data first?
    if (data_first) {
        price  = (const float*)d_in[0];
        volume = (const float*)d_in[1];
        adjs_p = (const float*)d_in[2];
        adjs_v = (const float*)d_in[3];
        par0 = 4;
    } else {
        // alphabetical top level: adjs_price, adjs_volume, params..., price, volume
        adjs_p = (const float*)d_in[0];
        adjs_v = (const float*)d_in[1];
        par0 = 2;
        price  = (const float*)d_in[n_in - 2];
        volume = (const float*)d_in[n_in - 1];
    }
    // insertion params start: Wp(2048), bp(64), ... ; sorted: Wp(2048), Wv(2048), ...
    const bool params_sorted = (in_sizes[par0 + 1] != 64);
    // canonical (insertion) index -> position in sorted layout
    static const int srt_map[30] = {
        0, 2, 1, 3,                         // Wp, bp, Wv, bv
        4, 5, 6, 7, 8, 9,                   // cca_p
        10, 11, 12, 13, 14, 15,             // cca_v
        27, 28, 29, 24, 25, 26, 23, 22,     // s: Wh,Wn,Wt,U,V,Vb,Rw,Rb
        18, 21, 16, 17, 19, 20              // g: W,w,U,Ub,u,ub
    };
    const float* prm[30];
    for (int i = 0; i < 30; ++i)
        prm[i] = (const float*)d_in[par0 + (params_sorted ? srt_map[i] : i)];

    const float *Wp = prm[0], *bp = prm[1], *Wv = prm[2], *bv = prm[3];
    const float *pW1 = prm[4], *pb1 = prm[5], *pW2 = prm[6], *pb2 = prm[7],
                *pW3 = prm[8], *pb3 = prm[9];
    const float *vW1 = prm[10], *vb1 = prm[11], *vW2 = prm[12], *vb2 = prm[13],
                *vW3 = prm[14], *vb3 = prm[15];
    const float *sWh = prm[16], *sWn = prm[17], *sWt = prm[18], *sU = prm[19],
                *sV = prm[20], *sVb = prm[21], *sRw = prm[22], *sRb = prm[23];
    const float *gW = prm[24], *gw = prm[25], *gU = prm[26], *gUb = prm[27],
                *gu = prm[28], *gub = prm[29];

    // ---- Workspace partition (fp32) ----
    float* wsp = (float*)d_ws;
    auto alloc = [&](size_t nf) { float* p = wsp; wsp += nf; return p; };
    float* x       = alloc((size_t)TT * NH);
    float* hpool[3]  = { alloc(NH), alloc(NH), alloc(NH) };
    float* cpool[3]  = { alloc(NH), alloc(NH), alloc(NH) };
    float* gpool[4]  = { alloc(NH), alloc(NH), alloc(NH), alloc(NH) };
    float* cgpool[4] = { alloc(NH), alloc(NH), alloc(NH), alloc(NH) };
    float* hn      = alloc(NH);
    float* support = alloc((size_t)RR * NH);
    float* Xcat    = alloc((size_t)NN * 320);
    float* gates   = alloc((size_t)NN * 320);
    float* abuf    = alloc(NH);
    float* bbuf    = alloc(NH);
    float* wtd     = alloc(NH);
    float* t1      = alloc((size_t)NN * 128);
    float* t2      = alloc((size_t)NN * 128);
    float* hps     = alloc(NH);          // h_p result
    float* Wcat    = alloc(320 * 320);   // [Wh|U|Wn|Wt|V]
    float* Wcat2   = alloc(320 * 128);   // [gW|gU]
    float* bias2   = alloc(320);

    float* out = (float*)d_out;          // [h_v | mlp_p | mlp_v]

    const dim3 B32(32);
    const dim3 PW_GRID(NH / 256);

    // ---- Build packed weights (step-invariant) ----
    pack5<<<dim3((320 * 320 + 255) / 256), 256, 0, stream>>>(sWh, sU, sWn, sWt, sV, Wcat, 320);
    pack2<<<dim3((320 * 128 + 255) / 256), 256, 0, stream>>>(gW, gU, Wcat2, 320);

    // ---- One branch of the temporal scan ----
    auto branch = [&](const float* adjs, float* result) {
        const float *h_t = x, *c_t = x, *g_t = x, *cg_t = x;  // carry init = x[0]
        int hi = 0, ci = 0, gi = 0, cgi = 0;
        for (int t = 0; t < TT; ++t) {
            const float* x_t = x + (size_t)t * NH;

            // glstm_init: gates = [g_t|h_t] @ [gW|gU].T + gUb
            float* g0  = gpool[gi];  gi = (gi + 1) & 3;
            float* cg0 = cgpool[cgi]; cgi = (cgi + 1) & 3;
            pack2<<<dim3((NN * 128 + 255) / 256), 256, 0, stream>>>(g_t, h_t, Xcat, NN);
            mm_wmma<<<dim3(NN / 16, 20), B32, 0, stream>>>(Xcat, Wcat2, gUb, gates,
                                                           128, 320, ACT_NONE);
            glstm_init_pw<<<PW_GRID, 256, 0, stream>>>(gates, cg_t, g0, cg0);

            const float *h_l = h_t, *c_l = c_t, *g_l = g0, *cg_l = cg0;
            for (int layer = 0; layer < 2; ++layer) {
                // RGCN: support[r] = h_l @ Rw[r].T + Rb[r] ; hn = tanh(sum_r A_r @ support[r])
                for (int r = 0; r < RR; ++r)
                    mm_wmma<<<dim3(NN / 16, HH / 16), B32, 0, stream>>>(
                        h_l, sRw + (size_t)r * HH * HH, sRb + r * HH,
                        support + (size_t)r * NH, HH, HH, ACT_NONE);
                rgcn_agg_wmma<<<dim3(NN / 16, HH / 16), B32, 0, stream>>>(adjs, support, hn);

                // SLSTM gates: [h|x|hn|h_t|g] @ [Wh|U|Wn|Wt|V].T + Vb
                pack5<<<dim3((NN * 320 + 255) / 256), 256, 0, stream>>>(
                    h_l, x_t, hn, h_t, g_l, Xcat, NN);
                mm_wmma<<<dim3(NN / 16, 20), B32, 0, stream>>>(Xcat, Wcat, sVb, gates,
                                                               320, 320, ACT_NONE);
                float* h_new = hpool[hi]; hi = (hi + 1) % 3;
                float* c_new = cpool[ci]; ci = (ci + 1) % 3;
                slstm_pw<<<PW_GRID, 256, 0, stream>>>(gates, c_l, h_t, h_new, c_new);

                // GLSTM: gates = sig(g_l @ gW.T + h_avg @ gU.T + gUb)
                havg_bias2<<<1, 320, 0, stream>>>(h_new, gU, gUb, bias2);
                mm_wmma<<<dim3(NN / 16, 20), B32, 0, stream>>>(g_l, gW, bias2, gates,
                                                               HH, 320, ACT_SIG);
                // a = g_l @ gw.T ; b = h_new @ gu.T + gub
                mm_wmma<<<dim3(NN / 16, HH / 16), B32, 0, stream>>>(g_l, gw, nullptr,
                                                                    abuf, HH, HH, ACT_NONE);
                mm_wmma<<<dim3(NN / 16, HH / 16), B32, 0, stream>>>(h_new, gu, gub,
                                                                    bbuf, HH, HH, ACT_NONE);
                fw_kernel<<<dim3(NN), dim3(HH), 0, stream>>>(abuf, bbuf, c_new, wtd);
                float* g_new  = gpool[gi];  gi = (gi + 1) & 3;
                float* cg_new = cgpool[cgi]; cgi = (cgi + 1) & 3;
                glstm_pw<<<PW_GRID, 256, 0, stream>>>(gates, cg_l, wtd, g_new, cg_new);

                h_l = h_new; c_l = c_new; g_l = g_new; cg_l = cg_new;
            }
            h_t = h_l; c_t = c_l; g_t = g_l; cg_t = cg_l;
        }
        copyk<<<PW_GRID, 256, 0, stream>>>(h_t, result, NH);
    };

    // ---- Price branch ----
    mm_wmma<<<dim3(TT * NN / 16, HH / 16), B32, 0, stream>>>(price, Wp, bp, x,
                                                             32, HH, ACT_NONE);
    branch(adjs_p, hps);

    // ---- Volume branch (reuses x) ----
    mm_wmma<<<dim3(TT * NN / 16, HH / 16), B32, 0, stream>>>(volume, Wv, bv, x,
                                                             32, HH, ACT_NONE);
    branch(adjs_v, out);                 // h_v -> out[0 : NH]

    // ---- MLP heads ----
    // mlp_p(h_p) -> out[NH : 2NH]
    mm_wmma<<<dim3(NN / 16, 128 / 16), B32, 0, stream>>>(hps, pW1, pb1, t1, HH, 128, ACT_RELU);
    mm_wmma<<<dim3(NN / 16, 128 / 16), B32, 0, stream>>>(t1, pW2, pb2, t2, 128, 128, ACT_RELU);
    mm_wmma<<<dim3(NN / 16, HH / 16),  B32, 0, stream>>>(t2, pW3, pb3, out + NH, 128, HH, ACT_NONE);
    // mlp_v(h_v) -> out[2NH : 3NH]
    mm_wmma<<<dim3(NN / 16, 128 / 16), B32, 0, stream>>>(out, vW1, vb1, t1, HH, 128, ACT_RELU);
    mm_wmma<<<dim3(NN / 16, 128 / 16), B32, 0, stream>>>(t1, vW2, vb2, t2, 128, 128, ACT_RELU);
    mm_wmma<<<dim3(NN / 16, HH / 16),  B32, 0, stream>>>(t2, vW3, vb3, out + 2 * NH, 128, HH, ACT_NONE);
}